// GCN_18408229830831
// MI455X (gfx1250) — compile-verified
//
#include <hip/hip_runtime.h>
#include <hip/hip_bf16.h>

typedef __attribute__((ext_vector_type(2))) float v2f;
typedef __attribute__((ext_vector_type(8))) float v8f;

// ---------------- degree / rsqrt ----------------
__global__ void deg_init_kernel(float* deg, int n) {
  int i = blockIdx.x * blockDim.x + threadIdx.x;
  if (i < n) deg[i] = 1.0f;  // self-loop contributes 1
}

__global__ void deg_edge_kernel(const int* __restrict__ col, float* deg, int E) {
  int e = blockIdx.x * blockDim.x + threadIdx.x;
  if (e < E) atomicAdd(&deg[col[e]], 1.0f);
}

__global__ void rsqrt_kernel(float* deg, int n) {
  int i = blockIdx.x * blockDim.x + threadIdx.x;
  if (i < n) deg[i] = rsqrtf(deg[i]);
}

// ---------------- GEMM: C[M,N] = act(A[M,K]) @ B[K,N] ----------------
// act(v) = relu(v + act_bias[k]) when BIAS_RELU (compile-time).
// N, K compile-time -> all loads use immediate offsets; no in-loop addressing.
// One wave per 16-wide N tile; wave's B tile (K x 16) preloaded into registers
// and reused across TM consecutive M tiles.
// f32 WMMA 16x16x4 layout: A lanes 0-15 hold K={k,k+1}, lanes 16-31 K={k+2,k+3};
// B mirrored (rows k..k+3, column n0+l); C/D: VGPR v -> rows v / v+8.
template <int N, int K, bool BIAS_RELU, int TM>
__global__ void gemm_wmma_kernel(const float* __restrict__ A,
                                 const float* __restrict__ B,
                                 float* __restrict__ C,
                                 const float* __restrict__ act_bias,
                                 int M) {
  const int lane = threadIdx.x & 31;
  const int wave = threadIdx.x >> 5;
  const int n0   = wave * 16;
  const int half = lane >> 4;   // 0: lanes 0-15, 1: lanes 16-31
  const int l    = lane & 15;
  const int kb   = 2 * half;    // this lane's k sub-offset within a 4-wide k step

  constexpr int KSTEPS = K / 4;

  // Preload this wave's B tile into registers (invariant across M tiles).
  v2f breg[KSTEPS];
  {
    const float* bcol = B + (size_t)kb * N + (n0 + l);
#pragma unroll
    for (int s = 0; s < KSTEPS; ++s) {
      breg[s].x = bcol[s * 4 * N];            // row kb + 4s
      breg[s].y = bcol[(s * 4 + 1) * N];      // row kb + 4s + 1
    }
  }
  const float* bias = act_bias + kb;

  for (int t = 0; t < TM; ++t) {              // uniform loop
    const int m0 = (blockIdx.x * TM + t) * 16;
    if (m0 >= M) break;                       // wave-uniform exit

    // clamp row so every lane stays active (EXEC must be all-ones for WMMA)
    int ar = m0 + l;
    if (ar > M - 1) ar = M - 1;
    const float* arow = A + (size_t)ar * K + kb;

    v8f acc = {};
#pragma unroll
    for (int s = 0; s < KSTEPS; ++s) {
      v2f a;
      a.x = arow[s * 4];
      a.y = arow[s * 4 + 1];
      if constexpr (BIAS_RELU) {
        a.x = fmaxf(a.x + bias[s * 4], 0.0f);     // loop-invariant: LICM hoists
        a.y = fmaxf(a.y + bias[s * 4 + 1], 0.0f);
      }
      acc = __builtin_amdgcn_wmma_f32_16x16x4_f32(false, a, false, breg[s],
                                                  (short)0, acc, false, false);
    }

    float* cp = C + (size_t)(m0 + half * 8) * N + (n0 + l);
#pragma unroll
    for (int v = 0; v < 8; ++v) {
      if (m0 + half * 8 + v < M) cp[(size_t)v * N] = acc[v];
    }
  }
}

// ---------------- self-loop + bias init: out = h*dinv^2 (+ bias) -------------
template <int F, bool HAS_BIAS>
__global__ void self_bias_init_kernel(const float* __restrict__ h,
                                      const float* __restrict__ dinv,
                                      const float* __restrict__ bias,
                                      float* __restrict__ out, int n) {
  int idx = blockIdx.x * blockDim.x + threadIdx.x;
  if (idx >= n * F) return;
  int i = idx / F;            // F is a power of two -> shift
  int f = idx - i * F;
  float d = dinv[i];
  float v = h[idx] * d * d;
  if constexpr (HAS_BIAS) v += bias[f];
  out[idx] = v;
}

// ---------------- edge aggregation: out[col] += h[row]*dinv[row]*dinv[col] ----
// One wave per edge; lane handles F/32 consecutive floats (float4 for F=128).
template <int F>
__global__ void edge_agg_kernel(const int* __restrict__ row,
                                const int* __restrict__ col,
                                const float* __restrict__ h,
                                const float* __restrict__ dinv,
                                float* out, int E) {
  const int lane = threadIdx.x & 31;
  const int e = blockIdx.x * (blockDim.x >> 5) + (threadIdx.x >> 5);
  if (e >= E) return;
  const int r = row[e];
  const int c = col[e];
  const float norm = dinv[r] * dinv[c];
  constexpr int VEC = F / 32;
  const float* hp = h + (size_t)r * F + lane * VEC;
  float* op = out + (size_t)c * F + lane * VEC;
  if constexpr (VEC == 4) {
    float4 hv = *(const float4*)hp;           // global_load_b128
    atomicAdd(op + 0, hv.x * norm);
    atomicAdd(op + 1, hv.y * norm);
    atomicAdd(op + 2, hv.z * norm);
    atomicAdd(op + 3, hv.w * norm);
  } else {
    float2 hv = *(const float2*)hp;           // global_load_b64
    atomicAdd(op + 0, hv.x * norm);
    atomicAdd(op + 1, hv.y * norm);
  }
}

extern "C" void kernel_launch(void* const* d_in, const int* in_sizes, int n_in,
                              void* d_out, int out_size, void* d_ws, size_t ws_size,
                              hipStream_t stream) {
  const float* x  = (const float*)d_in[0];
  const int*   ei = (const int*)d_in[1];
  const float* W1 = (const float*)d_in[2];
  const float* b1 = (const float*)d_in[3];
  const float* W2 = (const float*)d_in[4];
  const float* b2 = (const float*)d_in[5];

  const int hid    = in_sizes[3];            // 128
  const int out_ch = in_sizes[5];            // 64
  const int in_ch  = in_sizes[2] / hid;      // 128
  const int n      = in_sizes[0] / in_ch;    // 100000
  const int E      = in_sizes[1] / 2;        // 1.6M
  const int* row = ei;
  const int* col = ei + E;

  // workspace layout (fp32): dinv | h1 | agg1 | g2
  float* ws = (float*)d_ws;
  size_t off = 0;
  auto alloc = [&](size_t elems) {
    float* p = ws + off;
    off += (elems + 63) & ~(size_t)63;
    return p;
  };
  float* dinv = alloc((size_t)n);
  float* h1   = alloc((size_t)n * hid);
  float* agg1 = alloc((size_t)n * hid);
  float* g2   = alloc((size_t)n * out_ch);
  float* out  = (float*)d_out;

  // degrees -> dinv
  deg_init_kernel<<<(n + 255) / 256, 256, 0, stream>>>(dinv, n);
  deg_edge_kernel<<<(E + 255) / 256, 256, 0, stream>>>(col, dinv, E);
  rsqrt_kernel<<<(n + 255) / 256, 256, 0, stream>>>(dinv, n);

  constexpr int TM = 4;
  const int mtiles   = (n + 15) / 16;
  const int mblocks  = (mtiles + TM - 1) / TM;

  // layer 1: h1 = x @ W1   (N=128, K=128, no activation)
  gemm_wmma_kernel<128, 128, false, TM><<<mblocks, 32 * (128 / 16), 0, stream>>>(
      x, W1, h1, nullptr, n);
  // agg1 = h1*dinv^2 + sum_edges
  self_bias_init_kernel<128, false>
      <<<(unsigned)(((size_t)n * 128 + 255) / 256), 256, 0, stream>>>(
          h1, dinv, nullptr, agg1, n);
  edge_agg_kernel<128><<<(E + 7) / 8, 256, 0, stream>>>(row, col, h1, dinv, agg1, E);

  // layer 2: g2 = relu(agg1 + b1) @ W2   (bias+ReLU fused into A-load)
  gemm_wmma_kernel<64, 128, true, TM><<<mblocks, 32 * (64 / 16), 0, stream>>>(
      agg1, W2, g2, b1, n);
  // out = g2*dinv^2 + b2 + sum_edges
  self_bias_init_kernel<64, true>
      <<<(unsigned)(((size_t)n * 64 + 255) / 256), 256, 0, stream>>>(
          g2, dinv, b2, out, n);
  edge_agg_kernel<64><<<(E + 7) / 8, 256, 0, stream>>>(row, col, g2, dinv, out, E);
}